// NoisyTopkRouter_4312147165484
// MI455X (gfx1250) — compile-verified
//
#include <hip/hip_runtime.h>
#include <math.h>

// NoisyTopkRouter for MI455X (gfx1250, wave32).
// Memory-bound: x is 100.6MB fp32 read-once -> ~4.4us floor @ 23.3TB/s.
// Both 384->8 projections are fused into ONE fp32 WMMA chain:
//   A = combined weights [16 cols x K] (cols 0-7 route, 8-15 noise), from LDS
//   B = x tile           [K x 16 tokens], streamed from global
//   D[m=col][n=token] accumulated over K=384 via 96x v_wmma_f32_16x16x4_f32.
// D layout (32-bit C/D 16x16): lane l<16 holds token l's route logits in its
// 8 acc VGPRs; lane l+16 holds the SAME token's noise logits -> 8 shuffles
// fuse them, then per-lane softplus gate / top-2 / sparse softmax epilogue.

typedef float v2f __attribute__((ext_vector_type(2)));
typedef float v8f __attribute__((ext_vector_type(8)));

#define NDIM 384
#define WPAD 388   // LDS row stride in dwords; 388%64=4 -> conflict-free b64 reads

__global__ __launch_bounds__(128, 1) void noisy_topk_router_kernel(
    const float* __restrict__ x,
    const float* __restrict__ noise,
    const float* __restrict__ w_route,
    const float* __restrict__ b_route,
    const float* __restrict__ w_noise,
    const float* __restrict__ b_noise,
    float* __restrict__ out_probs,
    int*   __restrict__ out_idx,
    int n_groups)
{
    __shared__ float lw[16 * WPAD];   // combined weights, [col][k], padded

    const int tid  = threadIdx.x;
    const int wave = tid >> 5;
    const int lane = tid & 31;
    const int sub  = lane & 15;   // col (for A) / token-in-tile (for B)
    const int half = lane >> 4;   // K sub-offset selector
    const int koff = half * 2;

    // Stage combined weight matrix: col c<8 = w_route[c], c>=8 = w_noise[c-8].
    for (int i = tid; i < 16 * (NDIM / 4); i += 128) {
        const int c  = i / (NDIM / 4);
        const int k4 = i - c * (NDIM / 4);
        const float* src = (c < 8) ? (w_route + c * NDIM) : (w_noise + (c - 8) * NDIM);
        *(float4*)&lw[c * WPAD + k4 * 4] = *(const float4*)(src + k4 * 4);
    }
    __syncthreads();

    // A-fragment base for this lane: row M=sub (col of W), K elements {koff,koff+1}+4s
    const float* wrow = &lw[sub * WPAD + koff];

    for (int g = blockIdx.x; g < n_groups; g += gridDim.x) {
        const int t0 = (g * 4 + wave) * 16;             // this wave's 16-token tile
        // B-fragment base: column N=sub (token t0+sub), same K pattern
        const float* xrow = x + (size_t)(t0 + sub) * NDIM + koff;

        v8f acc = {};
        #pragma unroll 8
        for (int s = 0; s < NDIM / 4; ++s) {
            v2f a = *(const v2f*)(wrow + 4 * s);        // LDS  (weights)
            v2f b = *(const v2f*)(xrow + 4 * s);        // HBM  (activations, streamed)
            acc = __builtin_amdgcn_wmma_f32_16x16x4_f32(
                      false, a, false, b, (short)0, acc, false, false);
        }

        // Fuse halves: lanes<16 pull noise logits from lane sub+16.
        float nl[8];
        #pragma unroll
        for (int e = 0; e < 8; ++e)
            nl[e] = __shfl(acc[e], sub + 16, 32);

        if (half == 0) {
            const int t = t0 + sub;
            const float4 nz0 = *(const float4*)(noise + (size_t)t * 8);
            const float4 nz1 = *(const float4*)(noise + (size_t)t * 8 + 4);
            const float nzv[8] = {nz0.x, nz0.y, nz0.z, nz0.w,
                                  nz1.x, nz1.y, nz1.z, nz1.w};

            float best1 = -INFINITY, best2 = -INFINITY;
            int   i1 = 0, i2 = 0;
            float nv[8];
            #pragma unroll
            for (int e = 0; e < 8; ++e) {
                const float z  = nl[e] + b_noise[e];
                // stable softplus: max(z,0) + log1p(exp(-|z|))
                const float sp = fmaxf(z, 0.f) + log1pf(__expf(-fabsf(z)));
                const float v  = acc[e] + b_route[e] + nzv[e] * sp;
                nv[e] = v;
                if (v > best1)      { best2 = best1; i2 = i1; best1 = v; i1 = e; }
                else if (v > best2) { best2 = v; i2 = e; }   // strict '>' = JAX tie order
            }
            (void)nv;

            // softmax over the two kept logits; all others are exactly 0
            const float e2  = __expf(best2 - best1);
            const float inv = 1.f / (1.f + e2);
            float p[8];
            #pragma unroll
            for (int e = 0; e < 8; ++e)
                p[e] = (e == i1) ? inv : ((e == i2) ? e2 * inv : 0.f);

            *(float4*)(out_probs + (size_t)t * 8)     = make_float4(p[0], p[1], p[2], p[3]);
            *(float4*)(out_probs + (size_t)t * 8 + 4) = make_float4(p[4], p[5], p[6], p[7]);
            out_idx[(size_t)t * 2]     = i1;
            out_idx[(size_t)t * 2 + 1] = i2;
        }
        // branch reconverges here -> EXEC all-ones again before next WMMA chain
    }
}

extern "C" void kernel_launch(void* const* d_in, const int* in_sizes, int n_in,
                              void* d_out, int out_size, void* d_ws, size_t ws_size,
                              hipStream_t stream) {
    const float* x       = (const float*)d_in[0];
    const float* noise   = (const float*)d_in[1];
    const float* w_route = (const float*)d_in[2];
    const float* b_route = (const float*)d_in[3];
    const float* w_noise = (const float*)d_in[4];
    const float* b_noise = (const float*)d_in[5];

    const int n_tokens = in_sizes[0] / NDIM;   // 16*4096 = 65536
    const int n_groups = n_tokens / 64;        // 64 tokens (4 wave-tiles) per block-iter

    float* out_probs = (float*)d_out;                              // [B,T,8] f32
    int*   out_idx   = (int*)(out_probs + (size_t)n_tokens * 8);   // [B,T,2] i32

    const int grid = (n_groups < 512) ? n_groups : 512;
    noisy_topk_router_kernel<<<grid, 128, 0, stream>>>(
        x, noise, w_route, b_route, w_noise, b_noise, out_probs, out_idx, n_groups);
}